// CI_SFormer_85048942396025
// MI455X (gfx1250) — compile-verified
//
#include <hip/hip_runtime.h>

typedef float v2f __attribute__((ext_vector_type(2)));
typedef float v8f __attribute__((ext_vector_type(8)));
typedef unsigned int u32x4 __attribute__((ext_vector_type(4)));
typedef int i32x4 __attribute__((ext_vector_type(4)));
typedef int i32x8 __attribute__((ext_vector_type(8)));

// CDNA5 fp32 WMMA: D(16x16 f32) = A(16x4 f32) * B(4x16 f32) + C
static __device__ __forceinline__ v8f wmma_f32_16x16x4(v2f a, v2f b, v8f c) {
  return __builtin_amdgcn_wmma_f32_16x16x4_f32(false, a, false, b, (short)0, c, false, false);
}

#define NB   64
#define TT   128
#define VV   25
#define TV   3200   // T*V
#define CHUNK 4

// ---------- K0: combined weights  Wm2 = W2*Ws, bm2 = W2*bs + b2, row means ----------
__global__ void k_prep(const float* Wp, const float* bp, const float* Ws, const float* bs,
                       const float* W2, const float* b2,
                       float* Wm2, float* bm2, float* wpbar, float* wsbar, float* bbar) {
  int tid = threadIdx.x;
  for (int i = tid; i < 192 * 64; i += 256) {
    int o = i >> 6, ci = i & 63;
    float acc = 0.f;
    for (int c = 0; c < 64; ++c) acc += W2[o * 64 + c] * Ws[c * 64 + ci];
    Wm2[i] = acc;
  }
  for (int o = tid; o < 192; o += 256) {
    float acc = b2[o];
    for (int c = 0; c < 64; ++c) acc += W2[o * 64 + c] * bs[c];
    bm2[o] = acc;
  }
  if (tid < 64) {
    float a = 0.f; for (int c = 0; c < 8;  ++c) a += Wp[c * 64 + tid]; wpbar[tid] = a * 0.125f;
    float b = 0.f; for (int c = 0; c < 64; ++c) b += Ws[c * 64 + tid]; wsbar[tid] = b * (1.f / 64.f);
  }
  if (tid == 0) {
    float a = 0.f; for (int c = 0; c < 8;  ++c) a += bp[c]; bbar[0] = a * 0.125f;
    float b = 0.f; for (int c = 0; c < 64; ++c) b += bs[c]; bbar[1] = b * (1.f / 64.f);
  }
}

// ---------- K1: xtm[n,ci,v] = mean_t x[n,ci,t,v] ----------
__global__ void k_xtm(const float* x, float* xtm) {
  int blk = blockIdx.x;            // n*64+ci
  int v = threadIdx.x;
  if (v < VV) {
    const float* p = x + (size_t)blk * TV;
    float acc = 0.f;
    for (int t = 0; t < TT; ++t) acc += p[t * VV + v];
    xtm[blk * VV + v] = acc * (1.f / 128.f);
  }
}

// ---------- K2: A12 (6x25 softmax over q) and att2 (25x25 softmax over v) per n ----------
__global__ void k_stats1(const float* xtm, const float* wpbar, const float* wsbar, const float* bbar,
                         const float* w_t1, const float* w_t2,
                         const float* w_tw11, const float* w_tw21,
                         float* A12, float* att2) {
  const int J[6] = {3, 20, 7, 11, 14, 18};
  __shared__ float pS[25], sS[25], a1S[6], a2S[25], r1S[25], r2S[25];
  int n = blockIdx.x, tid = threadIdx.x;
  if (tid < VV) {
    float p = bbar[0], s = bbar[1];
    for (int ci = 0; ci < 64; ++ci) {
      float xv = xtm[(n * 64 + ci) * VV + tid];
      p += wpbar[ci] * xv;
      s += wsbar[ci] * xv;
    }
    pS[tid] = p; sS[tid] = s;
  }
  __syncthreads();
  if (tid < 6)  a1S[tid] = fmaxf(0.f, w_t1[0] * pS[J[tid]]);
  if (tid < VV) {
    a2S[tid] = fmaxf(0.f, w_t2[0] * pS[tid]);
    r1S[tid] = fmaxf(0.f, w_tw11[0] * sS[tid]);
    r2S[tid] = fmaxf(0.f, w_tw21[0] * sS[tid]);
  }
  __syncthreads();
  if (tid < VV) {
    // A12 column v=tid: softmax over q (6)
    float mx = -3.4e38f;
    for (int q = 0; q < 6; ++q) mx = fmaxf(mx, a1S[q] * a2S[tid]);
    float sum = 0.f;
    for (int q = 0; q < 6; ++q) sum += expf(a1S[q] * a2S[tid] - mx);
    for (int q = 0; q < 6; ++q) A12[(n * 6 + q) * VV + tid] = expf(a1S[q] * a2S[tid] - mx) / sum;
    // att2 column w=tid: softmax over v (25)
    float mx2 = -3.4e38f;
    for (int v = 0; v < VV; ++v) mx2 = fmaxf(mx2, r1S[v] * r2S[tid]);
    float sum2 = 0.f;
    for (int v = 0; v < VV; ++v) sum2 += expf(r1S[v] * r2S[tid] - mx2);
    for (int v = 0; v < VV; ++v) att2[(n * VV + v) * VV + tid] = expf(r1S[v] * r2S[tid] - mx2) / sum2;
  }
}

// ---------- K2b: x1_GCN[n,c,t,q] = sum_v (Wp@x+bp)[n,c,t,v] * A12[n,q,v] ----------
__global__ void k_x1gcn(const float* x, const float* Wp, const float* bp, const float* A12,
                        float* x1g) {
  __shared__ float WpS[512], bpS[8], AS[150], x1S[3200];
  int tb = blockIdx.x, n = blockIdx.y, tid = threadIdx.x;
  for (int i = tid; i < 512; i += 256) WpS[i] = Wp[i];
  if (tid < 8) bpS[tid] = bp[tid];
  for (int i = tid; i < 150; i += 256) AS[i] = A12[n * 150 + i];
  __syncthreads();
  for (int i = tid; i < 3200; i += 256) {     // 8c x 16t x 25v
    int c = i / 400, r = i % 400, t0 = r / 25, v = r % 25;
    int t = tb * 16 + t0;
    const float* xp = x + (size_t)n * 64 * TV + t * VV + v;   // stride TV per ci
    float acc = bpS[c];
    for (int ci = 0; ci < 64; ++ci) acc += WpS[c * 64 + ci] * xp[ci * TV];
    x1S[i] = acc;
  }
  __syncthreads();
  for (int i = tid; i < 768; i += 256) {      // 8c x 16t x 6q
    int c = i / 96, r = i % 96, t0 = r / 6, q = r % 6;
    float acc = 0.f;
    for (int v = 0; v < VV; ++v) acc += x1S[c * 400 + t0 * 25 + v] * AS[q * 25 + v];
    x1g[((size_t)(n * 8 + c) * TT + tb * 16 + t0) * 6 + q] = acc;
  }
}

// ---------- K3: xm -> att (6x6 softmax over v) per n ----------
__global__ void k_att(const float* x1g, const float* w_tw1, const float* w_tw2, float* att) {
  __shared__ float red[256 * 6];
  __shared__ float r1S[6], r2S[6];
  int n = blockIdx.x, tid = threadIdx.x;
  float acc[6] = {0, 0, 0, 0, 0, 0};
  for (int i = tid; i < 1024; i += 256) {     // (c,t) pairs
    const float* p = x1g + ((size_t)n * 1024 + i) * 6;
    for (int q = 0; q < 6; ++q) acc[q] += p[q];
  }
  for (int q = 0; q < 6; ++q) red[tid * 6 + q] = acc[q];
  __syncthreads();
  for (int s = 128; s >= 1; s >>= 1) {
    if (tid < s) for (int q = 0; q < 6; ++q) red[tid * 6 + q] += red[(tid + s) * 6 + q];
    __syncthreads();
  }
  if (tid < 6) {
    float xm = red[tid] * (1.f / 1024.f);
    r1S[tid] = fmaxf(0.f, w_tw1[0] * xm);
    r2S[tid] = fmaxf(0.f, w_tw2[0] * xm);
  }
  __syncthreads();
  if (tid < 6) {                               // column w: softmax over v
    float mx = -3.4e38f;
    for (int v = 0; v < 6; ++v) mx = fmaxf(mx, r1S[v] * r2S[tid]);
    float sum = 0.f;
    for (int v = 0; v < 6; ++v) sum += expf(r1S[v] * r2S[tid] - mx);
    for (int v = 0; v < 6; ++v) att[(n * 6 + v) * 6 + tid] = expf(r1S[v] * r2S[tid] - mx) / sum;
  }
}

// ---------- K4: y[n,c,t,w] = sum_k,v (W1@x1g+b1)[n,k*8+c,t,v]*(0.5*att+nlp) ----------
__global__ void k_y(const float* x1g, const float* W1, const float* b1, const float* att,
                    const float* Dpap, const float* pa, float* yb) {
  __shared__ float xS[6144], W1S[192], b1S[24], attS[36], nlpS[864];
  int n = blockIdx.x, tid = threadIdx.x;
  for (int i = tid; i < 6144; i += 256) xS[i] = x1g[(size_t)n * 6144 + i];
  for (int i = tid; i < 192; i += 256) W1S[i] = W1[i];
  if (tid < 24) b1S[tid] = b1[tid];
  for (int i = tid; i < 36; i += 256) attS[i] = att[n * 36 + i];
  for (int i = tid; i < 864; i += 256) nlpS[i] = Dpap[i] + pa[i];
  __syncthreads();
  for (int i = tid; i < 1024; i += 256) {
    int c = i / 128, t = i % 128;
    float m[3][6];
    for (int k = 0; k < 3; ++k) {
      int o = k * 8 + c;
      for (int v = 0; v < 6; ++v) {
        float acc = b1S[o];
        for (int cc = 0; cc < 8; ++cc) acc += W1S[o * 8 + cc] * xS[(cc * 128 + t) * 6 + v];
        m[k][v] = acc;
      }
    }
    for (int w = 0; w < 6; ++w) {
      float acc = 0.f;
      for (int k = 0; k < 3; ++k)
        for (int v = 0; v < 6; ++v)
          acc += m[k][v] * (0.5f * attS[v * 6 + w] + nlpS[(k * 8 + c) * 36 + v * 6 + w]);
      yb[((size_t)(n * 8 + c) * TT + t) * 6 + w] = acc;
    }
  }
}

// ---------- K5a: m2 chunk via WMMA; x-tile staged by Tensor Data Mover ----------
// C(192 x 32cols) = Wm2(192x64) * x-tile(64x32)
__global__ void k_m2(const float* x, const float* Wm2, const float* bm2, float* m2c, int n0) {
  extern __shared__ float sm[];
  float* WmS = sm;            // 192*64 = 12288
  float* XS  = sm + 12288;    // 64*32  = 2048
  int cn = blockIdx.y, n = n0 + cn;
  int colbase = blockIdx.x * 32;
  int tid = threadIdx.x;
  for (int i = tid; i < 12288; i += 256) WmS[i] = Wm2[i];

  // TDM 2D tile load: 64 rows (ci) x 32 cols, global row stride TV elems -> contiguous LDS.
  // D# per CDNA5 ISA (group0 + group1, groups 2/3 zero for 2D).
  if (tid < 32) {
    unsigned long long ga =
        (unsigned long long)(const void*)(x + (size_t)(n * 64) * TV + colbase);
    u32x4 g0;
    g0[0] = 1u;                                   // count=1, user mode, no gather
    g0[1] = (unsigned)(size_t)XS;                 // lds_addr (flat-shared addr[31:0] = LDS byte offset)
    g0[2] = (unsigned)(ga & 0xFFFFFFFFull);       // global_addr[31:0]
    g0[3] = (unsigned)((ga >> 32) & 0x01FFFFFFull) | (2u << 30);  // addr[56:32] | type=2
    i32x8 g1;
    g1[0] = (int)(2u << 16);                      // workgroup_mask=0, data_size=2 (4B)
    g1[1] = (int)(((unsigned)TV & 0xFFFFu) << 16);        // tensor_dim0[15:0]=3200
    g1[2] = (int)(((unsigned)TV >> 16) | (64u << 16));    // tensor_dim0[31:16] | tensor_dim1[15:0]=64
    g1[3] = (int)(32u << 16);                     // tensor_dim1[31:16] | tile_dim0=32
    g1[4] = 64;                                   // tile_dim1=64, tile_dim2=0
    g1[5] = TV;                                   // tensor_dim0_stride[31:0]=3200
    g1[6] = 0;                                    // stride0[47:32] | stride1[15:0]
    g1[7] = 0;                                    // stride1[47:16]
    i32x4 zz; zz[0] = 0; zz[1] = 0; zz[2] = 0; zz[3] = 0;
    i32x8 z8;
    for (int e = 0; e < 8; ++e) z8[e] = 0;
    // amdgpu-toolchain (clang-23) 6-arg form: (g0, g1, grp2, grp3, ext, cpol)
    __builtin_amdgcn_tensor_load_to_lds(g0, g1, zz, zz, z8, 0);
    __builtin_amdgcn_s_wait_tensorcnt(0);
  }
  __syncthreads();

  int wave = tid >> 5, lane = tid & 31, half = lane >> 4, l16 = lane & 15;
  v8f acc[3];
  for (int j = 0; j < 3; ++j) for (int e = 0; e < 8; ++e) acc[j][e] = 0.f;
  // 24 16x16 tiles (12 M x 2 N) over 8 waves -> 3 tiles/wave
  for (int k4 = 0; k4 < 64; k4 += 4) {
    for (int j = 0; j < 3; ++j) {
      int li = wave * 3 + j, mt = li >> 1, nt = li & 1;
      int row = mt * 16 + l16;
      v2f a; a.x = WmS[row * 64 + k4 + 2 * half];
             a.y = WmS[row * 64 + k4 + 2 * half + 1];
      int col = nt * 16 + l16;
      v2f b; b.x = XS[(k4 + 2 * half) * 32 + col];
             b.y = XS[(k4 + 2 * half + 1) * 32 + col];
      acc[j] = wmma_f32_16x16x4(a, b, acc[j]);
    }
  }
  for (int j = 0; j < 3; ++j) {
    int li = wave * 3 + j, mt = li >> 1, nt = li & 1;
    for (int r = 0; r < 8; ++r) {
      int M = mt * 16 + r + 8 * half;
      int Nc = colbase + nt * 16 + l16;
      m2c[(size_t)(cn * 192 + M) * TV + Nc] = acc[j][r] + bm2[M];
    }
  }
}

// ---------- K5b: y1[n,c] = sum_k m2_k(128x25) * (0.5*att2 + nlA_k)(25x25) via WMMA ----------
__global__ void k_y1(const float* m2c, const float* att2, const float* DecA, const float* A1b,
                     float* y1, int n0) {
  extern __shared__ float sm[];
  float* mS = sm;             // 3*128*28 = 10752 (v padded to 28, zero-filled)
  float* BS = sm + 10752;     // 3*28*32  = 2688  (w padded to 32)
  int c = blockIdx.x, cn = blockIdx.y, n = n0 + cn, g = c & 7;
  int tid = threadIdx.x;
  for (int i = tid; i < 10752; i += 256) {
    int k = i / 3584, r = i % 3584, t = r / 28, v = r % 28;
    mS[i] = (v < 25) ? m2c[(size_t)(cn * 192 + k * 64 + c) * TV + t * VV + v] : 0.f;
  }
  for (int i = tid; i < 2688; i += 256) {
    int k = i / 896, r = i % 896, v = r / 32, w = r & 31;
    float val = 0.f;
    if (v < 25 && w < 25) {
      int ai = ((k * 8 + g) * 25 + v) * 25 + w;
      val = 0.5f * att2[(n * VV + v) * VV + w] + DecA[ai] + A1b[ai];
    }
    BS[i] = val;
  }
  __syncthreads();
  int wave = tid >> 5, lane = tid & 31, half = lane >> 4, l16 = lane & 15;
  v8f acc0, acc1;
  for (int e = 0; e < 8; ++e) { acc0[e] = 0.f; acc1[e] = 0.f; }
  for (int k = 0; k < 3; ++k) {
    const float* mk = mS + k * 3584;
    const float* bk = BS + k * 896;
    for (int k4 = 0; k4 < 28; k4 += 4) {
      int row = wave * 16 + l16;                 // t row
      v2f a;  a.x  = mk[row * 28 + k4 + 2 * half];
              a.y  = mk[row * 28 + k4 + 2 * half + 1];
      v2f b0; b0.x = bk[(k4 + 2 * half) * 32 + l16];
              b0.y = bk[(k4 + 2 * half + 1) * 32 + l16];
      v2f b1; b1.x = bk[(k4 + 2 * half) * 32 + 16 + l16];
              b1.y = bk[(k4 + 2 * half + 1) * 32 + 16 + l16];
      acc0 = wmma_f32_16x16x4(a, b0, acc0);
      acc1 = wmma_f32_16x16x4(a, b1, acc1);
    }
  }
  for (int r = 0; r < 8; ++r) {
    int t = wave * 16 + r + 8 * half;
    size_t base = ((size_t)(n * 64 + c) * TT + t) * VV;
    y1[base + l16] = acc0[r];
    int w1 = 16 + l16;
    if (w1 < 25) y1[base + w1] = acc1[r];
  }
}

// ---------- K6a: a1l, a2l -> A12l (6x25, softmax over v) ----------
__global__ void k_a12l(const float* yb, const float* y1, const float* w_s1, const float* w_s2,
                       float* A12l) {
  __shared__ float red[256 * 25];
  __shared__ float a1lS[6], a2lS[25];
  int n = blockIdx.x, tid = threadIdx.x;
  {
    float acc[6] = {0, 0, 0, 0, 0, 0};
    for (int i = tid; i < 1024; i += 256) {
      const float* p = yb + ((size_t)n * 1024 + i) * 6;
      for (int q = 0; q < 6; ++q) acc[q] += p[q];
    }
    for (int q = 0; q < 6; ++q) red[tid * 6 + q] = acc[q];
    __syncthreads();
    for (int s = 128; s >= 1; s >>= 1) {
      if (tid < s) for (int q = 0; q < 6; ++q) red[tid * 6 + q] += red[(tid + s) * 6 + q];
      __syncthreads();
    }
    if (tid < 6) a1lS[tid] = fmaxf(0.f, w_s1[0] * red[tid] * (1.f / 1024.f));
    __syncthreads();
  }
  {
    float acc[25];
    for (int w = 0; w < 25; ++w) acc[w] = 0.f;
    for (int i = tid; i < 8192; i += 256) {
      const float* p = y1 + ((size_t)n * 8192 + i) * 25;
      for (int w = 0; w < 25; ++w) acc[w] += p[w];
    }
    for (int w = 0; w < 25; ++w) red[tid * 25 + w] = acc[w];
    __syncthreads();
    for (int s = 128; s >= 1; s >>= 1) {
      if (tid < s) for (int w = 0; w < 25; ++w) red[tid * 25 + w] += red[(tid + s) * 25 + w];
      __syncthreads();
    }
    if (tid < 25) a2lS[tid] = fmaxf(0.f, w_s2[0] * red[tid] * (1.f / 8192.f));
    __syncthreads();
  }
  if (tid < 25) {
    int q = tid;
    float mx = -3.4e38f;
    for (int v = 0; v < 6; ++v) mx = fmaxf(mx, a1lS[v] * a2lS[q]);
    float sum = 0.f;
    for (int v = 0; v < 6; ++v) sum += expf(a1lS[v] * a2lS[q] - mx);
    for (int v = 0; v < 6; ++v) A12l[(n * 6 + v) * 25 + q] = expf(a1lS[v] * a2lS[q] - mx) / sum;
  }
}

// ---------- K6b: y1 <- 0.5*(We @ (y x A12l) + be) + y1 ----------
__global__ void k_yl(const float* yb, const float* A12l, const float* We, const float* be,
                     float* y1) {
  __shared__ float yS[768], AS[150], WeS[512], beS[64], t1S[3200];
  int tb = blockIdx.x, n = blockIdx.y, tid = threadIdx.x;
  for (int i = tid; i < 768; i += 256) {
    int c = i / 96, r = i % 96, t0 = r / 6, v = r % 6;
    yS[i] = yb[((size_t)(n * 8 + c) * TT + tb * 16 + t0) * 6 + v];
  }
  for (int i = tid; i < 150; i += 256) AS[i] = A12l[n * 150 + i];
  for (int i = tid; i < 512; i += 256) WeS[i] = We[i];
  if (tid < 64) beS[tid] = be[tid];
  __syncthreads();
  for (int i = tid; i < 3200; i += 256) {     // t1[c,t0,q]
    int c = i / 400, r = i % 400, t0 = r / 25, q = r % 25;
    float acc = 0.f;
    for (int v = 0; v < 6; ++v) acc += yS[c * 96 + t0 * 6 + v] * AS[v * 25 + q];
    t1S[i] = acc;
  }
  __syncthreads();
  for (int i = tid; i < 25600; i += 256) {    // out[o,t0,q]
    int o = i / 400, r = i % 400, t0 = r / 25, q = r % 25;
    float acc = beS[o];
    for (int cc = 0; cc < 8; ++cc) acc += WeS[o * 8 + cc] * t1S[cc * 400 + t0 * 25 + q];
    size_t idx = ((size_t)(n * 64 + o) * TT + tb * 16 + t0) * VV + q;
    y1[idx] = 0.5f * acc + y1[idx];
  }
}

// ---------- K6c: per-(n,c) mean over (t,v) ----------
__global__ void k_qmean(const float* y1, float* qm) {
  __shared__ float red[128];
  int blk = blockIdx.x, tid = threadIdx.x;
  const float* p = y1 + (size_t)blk * TV;
  float acc = 0.f;
  for (int i = tid; i < TV; i += 128) acc += p[i];
  red[tid] = acc; __syncthreads();
  for (int s = 64; s >= 1; s >>= 1) { if (tid < s) red[tid] += red[tid + s]; __syncthreads(); }
  if (tid == 0) qm[blk] = red[0] * (1.f / 3200.f);
}

// ---------- K6d: channel attention scale + per-(n,c) BN partial sums ----------
__global__ void k_scale_sums(const float* y1, const float* qm, const float* k_ca,
                             float* sb, float* ps1, float* ps2) {
  __shared__ float r1[128], r2[128];
  int blk = blockIdx.x, tid = threadIdx.x;
  int c = blk & 63;
  float ql = (c > 0)  ? qm[blk - 1] : 0.f;
  float qr = (c < 63) ? qm[blk + 1] : 0.f;
  float qc = k_ca[0] * ql + k_ca[1] * qm[blk] + k_ca[2] * qr;
  float sv = 1.f + 1.f / (1.f + expf(-qc));
  const float* p = y1 + (size_t)blk * TV;
  float a1 = 0.f, a2 = 0.f;
  for (int i = tid; i < TV; i += 128) { float v = p[i] * sv; a1 += v; a2 += v * v; }
  r1[tid] = a1; r2[tid] = a2; __syncthreads();
  for (int s = 64; s >= 1; s >>= 1) {
    if (tid < s) { r1[tid] += r1[tid + s]; r2[tid] += r2[tid + s]; }
    __syncthreads();
  }
  if (tid == 0) { sb[blk] = sv; ps1[blk] = r1[0]; ps2[blk] = r2[0]; }
}

// ---------- K6e: per-channel mean / invstd (fixed-order over n: deterministic) ----------
__global__ void k_meanvar(const float* ps1, const float* ps2, float* mu, float* istd) {
  int c = threadIdx.x;
  if (c < 64) {
    float m1 = 0.f, m2 = 0.f;
    for (int n = 0; n < 64; ++n) { m1 += ps1[n * 64 + c]; m2 += ps2[n * 64 + c]; }
    const float inv = 1.f / (64.f * 3200.f);
    float mean = m1 * inv;
    float var = m2 * inv - mean * mean;
    mu[c] = mean;
    istd[c] = rsqrtf(var + 1e-5f);
  }
}

// ---------- K6f: out = relu(BN(y1*s) + x) ----------
__global__ void k_final(const float* y1, const float* sb, const float* mu, const float* istd,
                        const float* bn_g, const float* bn_b, const float* x, float* out) {
  size_t idx = (size_t)blockIdx.x * 256 + threadIdx.x;
  int nc = (int)(idx / TV);           // n*64+c
  int c = nc & 63;
  float v = y1[idx] * sb[nc];
  v = (v - mu[c]) * istd[c] * bn_g[c] + bn_b[c] + x[idx];
  out[idx] = fmaxf(v, 0.f);
}

extern "C" void kernel_launch(void* const* d_in, const int* in_sizes, int n_in,
                              void* d_out, int out_size, void* d_ws, size_t ws_size,
                              hipStream_t stream) {
  const float* x    = (const float*)d_in[0];
  const float* Dpap = (const float*)d_in[1];
  const float* pa   = (const float*)d_in[2];
  const float* DecA = (const float*)d_in[3];
  const float* A1b  = (const float*)d_in[4];
  const float* Wp   = (const float*)d_in[5];
  const float* bp   = (const float*)d_in[6];
  const float* Ws   = (const float*)d_in[7];
  const float* bs   = (const float*)d_in[8];
  const float* W1   = (const float*)d_in[9];
  const float* b1   = (const float*)d_in[10];
  const float* W2   = (const float*)d_in[11];
  const float* b2   = (const float*)d_in[12];
  const float* We   = (const float*)d_in[13];
  const float* be   = (const float*)d_in[14];
  const float* w_t1   = (const float*)d_in[15];
  const float* w_t2   = (const float*)d_in[16];
  const float* w_tw1  = (const float*)d_in[17];
  const float* w_tw2  = (const float*)d_in[18];
  const float* w_tw11 = (const float*)d_in[19];
  const float* w_tw21 = (const float*)d_in[20];
  const float* w_s1   = (const float*)d_in[21];
  const float* w_s2   = (const float*)d_in[22];
  const float* k_ca   = (const float*)d_in[23];
  const float* bn_g   = (const float*)d_in[24];
  const float* bn_b   = (const float*)d_in[25];
  float* out = (float*)d_out;

  float* wsb = (float*)d_ws;
  size_t off = 0;
  auto alloc = [&](size_t nf) { float* p = wsb + off; off += (nf + 15) & ~(size_t)15; return p; };
  float* Wm2   = alloc(192 * 64);
  float* bm2   = alloc(192);
  float* wpbar = alloc(64);
  float* wsbar = alloc(64);
  float* bbar  = alloc(16);
  float* xtm   = alloc(64 * 64 * 25);
  float* A12   = alloc(64 * 6 * 25);
  float* att2  = alloc(64 * 25 * 25);
  float* x1g   = alloc((size_t)64 * 8 * 128 * 6);
  float* att   = alloc(64 * 36);
  float* yb    = alloc((size_t)64 * 8 * 128 * 6);
  float* A12l  = alloc(64 * 6 * 25);
  float* qm    = alloc(64 * 64);
  float* sb    = alloc(64 * 64);
  float* ps1   = alloc(64 * 64);
  float* ps2   = alloc(64 * 64);
  float* mu    = alloc(64);
  float* istd  = alloc(64);
  float* m2c   = alloc((size_t)CHUNK * 192 * 3200);
  float* y1    = alloc((size_t)64 * 64 * 128 * 25);

  k_prep<<<1, 256, 0, stream>>>(Wp, bp, Ws, bs, W2, b2, Wm2, bm2, wpbar, wsbar, bbar);
  k_xtm<<<4096, 32, 0, stream>>>(x, xtm);
  k_stats1<<<64, 32, 0, stream>>>(xtm, wpbar, wsbar, bbar, w_t1, w_t2, w_tw11, w_tw21, A12, att2);
  k_x1gcn<<<dim3(8, 64), 256, 0, stream>>>(x, Wp, bp, A12, x1g);
  k_att<<<64, 256, 0, stream>>>(x1g, w_tw1, w_tw2, att);
  k_y<<<64, 256, 0, stream>>>(x1g, W1, b1, att, Dpap, pa, yb);
  for (int n0 = 0; n0 < 64; n0 += CHUNK) {
    k_m2<<<dim3(100, CHUNK), 256, (12288 + 2048) * 4, stream>>>(x, Wm2, bm2, m2c, n0);
    k_y1<<<dim3(64, CHUNK), 256, (10752 + 2688) * 4, stream>>>(m2c, att2, DecA, A1b, y1, n0);
  }
  k_a12l<<<64, 256, 0, stream>>>(yb, y1, w_s1, w_s2, A12l);
  k_yl<<<dim3(8, 64), 256, 0, stream>>>(yb, A12l, We, be, y1);
  k_qmean<<<4096, 128, 0, stream>>>(y1, qm);
  k_scale_sums<<<4096, 128, 0, stream>>>(y1, qm, k_ca, sb, ps1, ps2);
  k_meanvar<<<1, 64, 0, stream>>>(ps1, ps2, mu, istd);
  k_final<<<51200, 256, 0, stream>>>(y1, sb, mu, istd, bn_g, bn_b, x, out);

  (void)in_sizes; (void)n_in; (void)out_size; (void)ws_size;
}